// GCNNodeRegressor_67044439491025
// MI455X (gfx1250) — compile-verified
//
#include <hip/hip_runtime.h>
#include <hip/hip_bf16.h>

// GCN (3x GCNConv + linear readout) for MI455X / gfx1250.
// Memory-bound problem; edge gather/scatter working set (~115 MB) fits in the
// 192 MB L2, so float-atomic scatter is the right segment_sum. GEMMs use the
// full-precision f32 matrix path V_WMMA_F32_16X16X4_F32 (wave32).

typedef __attribute__((ext_vector_type(2))) float v2f;
typedef __attribute__((ext_vector_type(8))) float v8f;

#define HID 64  // hidden width fixed by the reference

// ---------------------------------------------------------------------------
// GEMM: C[M x 64] = A[M x K] * B[K x 64], row-major, f32, via WMMA 16x16x4.
// One wave computes a 16-row x 64-col slab (4 N-tiles of 16).
// A fragment (16x4 f32, 2 VGPRs): lane l: m = l&15, kh = l>>4;
//   a[0] = A[m][k+2*kh], a[1] = A[m][k+2*kh+1]   (per ISA 7.12.2 table)
// B fragment (4x16 f32, 2 VGPRs): lane l: n = l&15, kh = l>>4;
//   b[0] = B[k+2*kh][n], b[1] = B[k+2*kh+1][n]
// C/D (16x16 f32, 8 VGPRs): element v of lane l = C[row0 + v + 8*(l>>4)][l&15]
// ---------------------------------------------------------------------------
__global__ void __launch_bounds__(256)
gemm_wmma_f32(const float* __restrict__ A, const float* __restrict__ B,
              float* __restrict__ C, int M, int K) {
  extern __shared__ float sB[];  // K * 64 floats

  // Cooperative stage of B into LDS (K*64*4 bytes: 32 KB max, << 320 KB WGP LDS)
  const int total = K * HID;
  for (int i = threadIdx.x; i < total; i += 256) sB[i] = B[i];
  __syncthreads();

  const int lane = threadIdx.x & 31;
  const int wave = blockIdx.x * 8 + (threadIdx.x >> 5);
  const int row0 = wave * 16;
  if (row0 >= M) return;  // wave-uniform: EXEC stays all-1s for WMMA

  const int m  = lane & 15;  // A row within tile; doubles as B/C column index
  const int kh = lane >> 4;  // which K-pair this half-wave holds

  v8f c0 = {}, c1 = {}, c2 = {}, c3 = {};
  const float* __restrict__ arow = A + (size_t)(row0 + m) * K + 2 * kh;

  for (int k = 0; k < K; k += 4) {
    v2f a = *(const v2f*)(arow + k);  // 8B aligned (k+2*kh even)
    const float* bp = sB + (k + 2 * kh) * HID + m;
    v2f b0 = { bp[0],        bp[HID + 0]  };
    v2f b1 = { bp[16],       bp[HID + 16] };
    v2f b2 = { bp[32],       bp[HID + 32] };
    v2f b3 = { bp[48],       bp[HID + 48] };
    c0 = __builtin_amdgcn_wmma_f32_16x16x4_f32(false, a, false, b0, (short)0, c0, false, false);
    c1 = __builtin_amdgcn_wmma_f32_16x16x4_f32(false, a, false, b1, (short)0, c1, false, false);
    c2 = __builtin_amdgcn_wmma_f32_16x16x4_f32(false, a, false, b2, (short)0, c2, false, false);
    c3 = __builtin_amdgcn_wmma_f32_16x16x4_f32(false, a, false, b3, (short)0, c3, false, false);
  }

  // Store the 16x64 slab.
  #pragma unroll
  for (int v = 0; v < 8; ++v) {
    size_t r = (size_t)(row0 + v + 8 * kh) * HID + m;
    C[r +  0] = c0[v];
    C[r + 16] = c1[v];
    C[r + 32] = c2[v];
    C[r + 48] = c3[v];
  }
}

// ---------------------------------------------------------------------------
__global__ void zero_f32(float* __restrict__ p, int n) {
  int i = blockIdx.x * blockDim.x + threadIdx.x;
  if (i < n) p[i] = 0.0f;
}

__global__ void deg_count(const int* __restrict__ dst, float* __restrict__ deg, int E) {
  int e = blockIdx.x * blockDim.x + threadIdx.x;
  if (e < E) atomicAdd(&deg[dst[e]], 1.0f);
}

// deg includes the self-loop (+1), so deg >= 1 always -> no zero guard needed.
__global__ void deg_finalize(float* __restrict__ deg, int n) {
  int i = blockIdx.x * blockDim.x + threadIdx.x;
  if (i < n) deg[i] = rsqrtf(deg[i] + 1.0f);
}

// out[i][c] = hw[i][c] * dis[i]^2   (the self-loop message)
__global__ void self_loop_init(const float* __restrict__ hw, const float* __restrict__ dis,
                               float* __restrict__ out, int total) {
  int idx = blockIdx.x * blockDim.x + threadIdx.x;
  if (idx < total) {
    float d = dis[idx >> 6];
    out[idx] = hw[idx] * d * d;
  }
}

// For each edge e and channel c: out[dst][c] += hw[src][c] * dis[src]*dis[dst].
// 64 consecutive threads share one edge -> fully coalesced gather and atomics;
// dis[] lookups hit L0. Atomics land in L2 (working set < 192 MB).
__global__ void edge_scatter(const float* __restrict__ hw, const float* __restrict__ dis,
                             const int* __restrict__ src, const int* __restrict__ dst,
                             float* __restrict__ out, int E) {
  int idx = blockIdx.x * blockDim.x + threadIdx.x;  // E*64 = 102.4M < 2^31
  int e = idx >> 6;
  if (e >= E) return;
  int c = idx & 63;
  int s = src[e], d = dst[e];
  float w = dis[s] * dis[d];
  atomicAdd(&out[(size_t)d * HID + c], hw[(size_t)s * HID + c] * w);
}

__global__ void bias_relu(float* __restrict__ h, const float* __restrict__ b, int total) {
  int idx = blockIdx.x * blockDim.x + threadIdx.x;
  if (idx < total) {
    float v = h[idx] + b[idx & 63];
    h[idx] = v > 0.0f ? v : 0.0f;
  }
}

__global__ void readout_kernel(const float* __restrict__ h, const float* __restrict__ w,
                               const float* __restrict__ bo, float* __restrict__ out, int n) {
  int i = blockIdx.x * blockDim.x + threadIdx.x;
  if (i >= n) return;
  const float* hr = h + (size_t)i * HID;
  float s = bo[0];
  #pragma unroll
  for (int c = 0; c < HID; ++c) s = fmaf(hr[c], w[c], s);
  out[i] = s;
}

// ---------------------------------------------------------------------------
extern "C" void kernel_launch(void* const* d_in, const int* in_sizes, int n_in,
                              void* d_out, int out_size, void* d_ws, size_t ws_size,
                              hipStream_t stream) {
  const float* x  = (const float*)d_in[0];
  const int*   ei = (const int*)d_in[1];   // edge_index, [2, E] flattened
  const float* W1 = (const float*)d_in[2];
  const float* b1 = (const float*)d_in[3];
  const float* W2 = (const float*)d_in[4];
  const float* b2 = (const float*)d_in[5];
  const float* W3 = (const float*)d_in[6];
  const float* b3 = (const float*)d_in[7];
  const float* Wo = (const float*)d_in[8];
  const float* bo = (const float*)d_in[9];
  float* out = (float*)d_out;

  const int N  = out_size;            // 100000
  const int E  = in_sizes[1] / 2;     // 1600000
  const int K1 = in_sizes[0] / N;     // 128 (input channels)

  const int* src = ei;
  const int* dst = ei + E;

  // Workspace carve: dis[N] | buf0[N*64] | buf1[N*64]  (~51.6 MB)
  float* dis  = (float*)d_ws;
  float* buf0 = dis + (((size_t)N + 63) & ~(size_t)63);
  float* buf1 = buf0 + (size_t)N * HID;

  const int TB = 256;
  const int totalH = N * HID;
  const long long ew = (long long)E * HID;

  // --- normalization: deg_inv_sqrt ---
  zero_f32     <<<(N + TB - 1) / TB, TB, 0, stream>>>(dis, N);
  deg_count    <<<(E + TB - 1) / TB, TB, 0, stream>>>(dst, dis, E);
  deg_finalize <<<(N + TB - 1) / TB, TB, 0, stream>>>(dis, N);

  auto layer = [&](const float* hin, int K, const float* W, const float* b) {
    int waves  = (N + 15) / 16;
    int blocks = (waves + 7) / 8;
    gemm_wmma_f32 <<<blocks, TB, (size_t)K * HID * sizeof(float), stream>>>(hin, W, buf0, N, K);
    self_loop_init<<<(totalH + TB - 1) / TB, TB, 0, stream>>>(buf0, dis, buf1, totalH);
    edge_scatter  <<<(int)((ew + TB - 1) / TB), TB, 0, stream>>>(buf0, dis, src, dst, buf1, E);
    bias_relu     <<<(totalH + TB - 1) / TB, TB, 0, stream>>>(buf1, b, totalH);
  };

  layer(x,    K1,  W1, b1);   // 128 -> 64
  layer(buf1, HID, W2, b2);   //  64 -> 64
  layer(buf1, HID, W3, b3);   //  64 -> 64

  readout_kernel<<<(N + TB - 1) / TB, TB, 0, stream>>>(buf1, Wo, bo, out, N);
}